// RGCNLayer_3693671875022
// MI455X (gfx1250) — compile-verified
//
#include <hip/hip_runtime.h>
#include <hip/hip_bf16.h>

typedef float v2f __attribute__((ext_vector_type(2)));
typedef float v8f __attribute__((ext_vector_type(8)));

#define N_NODES 100000
#define N_EDGES 1000000
#define DIM     64
#define N_REL   4

// Workspace layout (in floats):
//   h[5][N_NODES][64]   : 5 * 6,400,000 = 32,000,000   (slot 4 = x @ root)
//   agg[N_NODES][64]    : 6,400,000                     (offset 32,000,000)
//   cnt[N_NODES]        : 100,000                       (offset 38,400,000)
#define H_OFF    0
#define AGG_OFF  32000000
#define CNT_OFF  38400000
#define ZERO_CNT 6500000   // agg + cnt are contiguous

// ---------------------------------------------------------------------------
// Kernel 1: h[m] = x @ W_m (m<4) or x @ root (m==4), fp32 WMMA 16x16x4.
// Block = 128 threads (4 waves). blockIdx.x = node tile (16 rows),
// blockIdx.y = matrix index m in [0,5). Each wave owns one 16-col tile.
// ---------------------------------------------------------------------------
__global__ __launch_bounds__(128) void rgcn_transform_kernel(
    const float* __restrict__ x,
    const float* __restrict__ weight,   // [4][64][64]
    const float* __restrict__ root,     // [64][64]
    float* __restrict__ ws)
{
    __shared__ float xs[16 * 66];       // stride 66: conflict-free, 8B aligned

    const int tile    = blockIdx.x;
    const int m       = blockIdx.y;
    const int rowbase = tile * 16;
    const float* __restrict__ W = (m < 4) ? (weight + m * DIM * DIM) : root;
    float* __restrict__ hout = ws + H_OFF + (size_t)m * N_NODES * DIM;

    // Stage 16x64 x-tile into LDS: each of 128 threads loads 8 floats.
    {
        const int t = threadIdx.x;
        const int r = t >> 3;
        const int c = (t & 7) * 8;
        const float* src = x + (size_t)(rowbase + r) * DIM + c;
        float4 p0 = *(const float4*)(src);
        float4 p1 = *(const float4*)(src + 4);
        float* d = xs + r * 66 + c;
        d[0] = p0.x; d[1] = p0.y; d[2] = p0.z; d[3] = p0.w;
        d[4] = p1.x; d[5] = p1.y; d[6] = p1.z; d[7] = p1.w;
    }
    __syncthreads();

    const int wave    = threadIdx.x >> 5;    // 0..3 -> output column tile
    const int lane    = threadIdx.x & 31;
    const int colbase = wave * 16;
    const int half    = lane >> 4;           // 0: K={0,1}, 1: K={2,3}
    const int mn      = lane & 15;           // M for A frag, N for B frag
    const int kb      = half * 2;

    v8f acc = {0.f, 0.f, 0.f, 0.f, 0.f, 0.f, 0.f, 0.f};

#pragma unroll
    for (int kk = 0; kk < DIM; kk += 4) {
        v2f a, b;
        const float* ap = xs + mn * 66 + kk + kb;      // A[M][kk+kb..+1]
        a.x = ap[0];
        a.y = ap[1];
        const float* bp = W + (size_t)(kk + kb) * DIM + colbase + mn;  // B[k][N]
        b.x = bp[0];
        b.y = bp[DIM];
        // 8 args: (neg_a, A, neg_b, B, c_mod, C, reuse_a, reuse_b)
        acc = __builtin_amdgcn_wmma_f32_16x16x4_f32(
            false, a, false, b, (short)0, acc, false, false);
    }

    // C/D layout: VGPR v -> row (v + 8*half), col = lane&15.
#pragma unroll
    for (int v = 0; v < 8; ++v) {
        const int r = v + half * 8;
        hout[(size_t)(rowbase + r) * DIM + colbase + mn] = acc[v];
    }
}

// ---------------------------------------------------------------------------
// Kernel 2: zero agg + cnt regions of workspace.
// ---------------------------------------------------------------------------
__global__ __launch_bounds__(256) void rgcn_zero_kernel(float* __restrict__ ws)
{
    const int i = blockIdx.x * 256 + threadIdx.x;
    if (i < ZERO_CNT) ws[AGG_OFF + i] = 0.0f;
}

// ---------------------------------------------------------------------------
// Kernel 3: one wave per edge. Gather h[etype][src][0..63] (L2-resident),
// scatter-add into agg[dst], bump in-degree counter.
// ---------------------------------------------------------------------------
__global__ __launch_bounds__(256) void rgcn_edge_kernel(
    const long long* __restrict__ edge_index,  // [2][N_EDGES]
    const long long* __restrict__ edge_type,   // [N_EDGES]
    float* __restrict__ ws)
{
    const int e = blockIdx.x * 8 + (threadIdx.x >> 5);
    const int lane = threadIdx.x & 31;

    const long long src = edge_index[e];
    const long long dst = edge_index[N_EDGES + e];
    const long long rel = edge_type[e];

    const float* __restrict__ hrow =
        ws + H_OFF + ((size_t)rel * N_NODES + (size_t)src) * DIM;
    float* __restrict__ arow = ws + AGG_OFF + (size_t)dst * DIM;

    const float m0 = hrow[lane];
    const float m1 = hrow[lane + 32];
    atomicAdd(&arow[lane],      m0);
    atomicAdd(&arow[lane + 32], m1);
    if (lane == 0) atomicAdd(&ws[CNT_OFF + dst], 1.0f);
}

// ---------------------------------------------------------------------------
// Kernel 4: out = tanh(agg / max(cnt,1) + x@root + bias)
// ---------------------------------------------------------------------------
__global__ __launch_bounds__(256) void rgcn_final_kernel(
    const float* __restrict__ ws,
    const float* __restrict__ bias,
    float* __restrict__ out)
{
    const int i = blockIdx.x * 256 + threadIdx.x;   // over N_NODES*64
    const int node = i >> 6;
    const int c    = i & 63;

    const float a  = ws[AGG_OFF + i];
    const float cn = ws[CNT_OFF + node];
    const float hr = ws[H_OFF + (size_t)4 * N_NODES * DIM + i];  // x @ root
    const float o  = a / fmaxf(cn, 1.0f) + hr + bias[c];
    out[i] = tanhf(o);
}

// ---------------------------------------------------------------------------
extern "C" void kernel_launch(void* const* d_in, const int* in_sizes, int n_in,
                              void* d_out, int out_size, void* d_ws, size_t ws_size,
                              hipStream_t stream) {
    const float*     x          = (const float*)d_in[0];
    const long long* edge_index = (const long long*)d_in[1];
    const long long* edge_type  = (const long long*)d_in[2];
    const float*     weight     = (const float*)d_in[3];
    const float*     root       = (const float*)d_in[4];
    const float*     bias       = (const float*)d_in[5];
    float*           out        = (float*)d_out;
    float*           ws         = (float*)d_ws;

    // 1) dense transforms via fp32 WMMA
    dim3 gT(N_NODES / 16, 5);
    rgcn_transform_kernel<<<gT, 128, 0, stream>>>(x, weight, root, ws);

    // 2) clear accumulators
    rgcn_zero_kernel<<<(ZERO_CNT + 255) / 256, 256, 0, stream>>>(ws);

    // 3) gather + mean-scatter over edges
    rgcn_edge_kernel<<<N_EDGES / 8, 256, 0, stream>>>(edge_index, edge_type, ws);

    // 4) epilogue
    rgcn_final_kernel<<<(N_NODES * DIM) / 256, 256, 0, stream>>>(ws, bias, out);
}